// Geo_Loss_19533511262814
// MI455X (gfx1250) — compile-verified
//
#include <hip/hip_runtime.h>
#include <math.h>

typedef float v2f __attribute__((ext_vector_type(2)));
typedef float v8f __attribute__((ext_vector_type(8)));

#define NBATCH 8192
#define NPTS   1024

// ---------------- kernel 0: zero the 3 output scalars ----------------
__global__ void gl_zero(float* out) {
    if (threadIdx.x < 3) out[threadIdx.x] = 0.0f;
}

// Build E = [R[:,0] | R[:,1] | R@t] (row-major 3x3) from quat+translation,
// following the reference's quat->eulerXYZ->R(=transpose(r1 r2 r3)) path.
__device__ __forceinline__ void quat_to_E(float q0, float q1, float q2, float q3,
                                          float t0, float t1, float t2,
                                          float* __restrict__ E) {
    float ax = atan2f(2.0f * (q0 * q1 + q2 * q3), 1.0f - 2.0f * (q1 * q1 + q2 * q2));
    float s  = 2.0f * (q0 * q2 - q1 * q3);
    s = fminf(1.0f, fmaxf(-1.0f, s));
    float ay = asinf(s);
    float az = atan2f(2.0f * (q0 * q3 + q1 * q2), 1.0f - 2.0f * (q2 * q2 + q3 * q3));
    float sx, cx, sy, cy, sz, cz;
    sincosf(ax, &sx, &cx);
    sincosf(ay, &sy, &cy);
    sincosf(az, &sz, &cz);
    // R = (r1 r2 r3)^T
    float R00 = cy * cz, R01 = sx * sy * cz - cx * sz, R02 = cx * sy * cz + sx * sz;
    float R10 = cy * sz, R11 = sx * sy * sz + cx * cz, R12 = cx * sy * sz - sx * cz;
    float R20 = -sy,     R21 = sx * cy,                R22 = cx * cy;
    float rt0 = R00 * t0 + R01 * t1 + R02 * t2;
    float rt1 = R10 * t0 + R11 * t1 + R12 * t2;
    float rt2 = R20 * t0 + R21 * t1 + R22 * t2;
    E[0] = R00; E[1] = R01; E[2] = rt0;
    E[3] = R10; E[4] = R11; E[5] = rt1;
    E[6] = R20; E[7] = R21; E[8] = rt2;
}

// ---------------- kernel 1: per-batch extrinsics + L1 losses ----------------
__global__ __launch_bounds__(256)
void gl_extrin(const float* __restrict__ y, const float* __restrict__ y_pred,
               float* __restrict__ ws, float* __restrict__ out) {
    __shared__ float red[16];
    int b = blockIdx.x * blockDim.x + threadIdx.x;
    float lt = 0.0f, lr = 0.0f;
    if (b < NBATCH) {
        const float* yb = y + b * 7;
        const float* pb = y_pred + b * 5;
        lt = fabsf(pb[0] - yb[2]);  // yp[:, :2] == y[:, :2]
        lr = fabsf(pb[1] - yb[3]) + fabsf(pb[2] - yb[4]) +
             fabsf(pb[3] - yb[5]) + fabsf(pb[4] - yb[6]);
        // predicted pose (normalized quat); v_rsq_f32
        float q0 = pb[1], q1 = pb[2], q2 = pb[3], q3 = pb[4];
        float inq = __builtin_amdgcn_rsqf(q0 * q0 + q1 * q1 + q2 * q2 + q3 * q3);
        float Ep[9], El[9];
        quat_to_E(q0 * inq, q1 * inq, q2 * inq, q3 * inq, yb[0], yb[1], pb[0], Ep);
        quat_to_E(yb[3], yb[4], yb[5], yb[6], yb[0], yb[1], yb[2], El);
        float* w = ws + b * 18;
        #pragma unroll
        for (int i = 0; i < 9; ++i) { w[i] = Ep[i]; w[9 + i] = El[i]; }
    }
    // block reduction of the two L1 terms (wave32)
    #pragma unroll
    for (int o = 16; o > 0; o >>= 1) {
        lt += __shfl_down(lt, o, 32);
        lr += __shfl_down(lr, o, 32);
    }
    int wid = threadIdx.x >> 5;
    if ((threadIdx.x & 31) == 0) { red[wid] = lt; red[8 + wid] = lr; }
    __syncthreads();
    if (threadIdx.x == 0) {
        float st = 0.0f, sr = 0.0f;
        #pragma unroll
        for (int i = 0; i < 8; ++i) { st += red[i]; sr += red[8 + i]; }
        atomicAdd(out + 1, st * (1.0f / (NBATCH * 3.0f)));
        atomicAdd(out + 2, sr * (1.0f / (NBATCH * 4.0f)));
    }
}

// 3x3 inverse via adjugate (v_rcp_f32 for the determinant)
__device__ __forceinline__ void inv3(const float* __restrict__ M, float* __restrict__ I) {
    float a00 = M[4] * M[8] - M[5] * M[7];
    float a01 = M[2] * M[7] - M[1] * M[8];
    float a02 = M[1] * M[5] - M[2] * M[4];
    float a10 = M[5] * M[6] - M[3] * M[8];
    float a11 = M[0] * M[8] - M[2] * M[6];
    float a12 = M[2] * M[3] - M[0] * M[5];
    float a20 = M[3] * M[7] - M[4] * M[6];
    float a21 = M[1] * M[6] - M[0] * M[7];
    float a22 = M[0] * M[4] - M[1] * M[3];
    float r = __builtin_amdgcn_rcpf(M[0] * a00 + M[1] * a10 + M[2] * a20);
    I[0] = a00 * r; I[1] = a01 * r; I[2] = a02 * r;
    I[3] = a10 * r; I[4] = a11 * r; I[5] = a12 * r;
    I[6] = a20 * r; I[7] = a21 * r; I[8] = a22 * r;
}

__device__ __forceinline__ float point_term(float x, float yv,
                                            const float* __restrict__ P,
                                            const float* __restrict__ L) {
    bool m = (x > 0.0f) && (yv > 0.0f);
    float ppx = fmaf(P[0], x, fmaf(P[1], yv, P[2]));
    float ppy = fmaf(P[3], x, fmaf(P[4], yv, P[5]));
    float ppz = fmaf(P[6], x, fmaf(P[7], yv, P[8]));
    float plx = fmaf(L[0], x, fmaf(L[1], yv, L[2]));
    float ply = fmaf(L[3], x, fmaf(L[4], yv, L[5]));
    float plz = fmaf(L[6], x, fmaf(L[7], yv, L[8]));
    float zp = m ? ppz : 1.0f;
    float zl = m ? plz : 1.0f;
    float r  = __builtin_amdgcn_rcpf(zp * zl);  // v_rcp_f32: one reciprocal for both divides
    float dx = fmaf(ppx, zl, -plx * zp);        // (ppx/zp - plx/zl) * (zp*zl)
    float dy = fmaf(ppy, zl, -ply * zp);
    float ss = fmaf(dx, dx, dy * dy);
    float nrm = __builtin_amdgcn_sqrtf(ss) * fabsf(r);  // v_sqrt_f32
    return m ? nrm : 0.0f;
}

// ---------------- kernel 2: WMMA cam3 = K@E, invert, stream X ----------------
__global__ __launch_bounds__(256)
void gl_main(const float* __restrict__ X, const float* __restrict__ ws,
             const float* __restrict__ Kin, float* __restrict__ out) {
    __shared__ float lds_cam[18];
    __shared__ float red[8];
    const int b   = blockIdx.x;
    const int tid = threadIdx.x;
    const int lane = tid & 31;
    const bool hi  = lane >= 16;
    const int n    = hi ? lane - 16 : lane;

    // A-matrix (16x4 f32): rows 0-2 = intrinsics K, K-dim padded with 0.
    // ISA layout: lanes 0-15 hold (K=0,K=1), lanes 16-31 hold (K=2,K=3).
    float a0 = 0.0f, a1 = 0.0f;
    if (!hi) {
        if (lane < 3) { a0 = Kin[lane * 3 + 0]; a1 = Kin[lane * 3 + 1]; }
    } else {
        if (n < 3)    { a0 = Kin[n * 3 + 2]; }
    }
    // B-matrix (4x16 f32): N-cols 0-2 = E_pred cols, 3-5 = E_lab cols, rest 0.
    // Layout: VGPR0 = rows k=0 (lanes 0-15) / k=2 (lanes 16-31), VGPR1 = k=1 / k=3.
    float b0 = 0.0f, b1 = 0.0f;
    if (n < 6) {
        const float* E = ws + b * 18 + (n < 3 ? 0 : 9);
        int col = (n < 3) ? n : n - 3;
        if (!hi) { b0 = E[0 + col]; b1 = E[3 + col]; }
        else     { b0 = E[6 + col]; /* k=3 row is zero */ }
    }
    v2f Am = {a0, a1};
    v2f Bm = {b0, b1};
    v8f Cm = {0.f, 0.f, 0.f, 0.f, 0.f, 0.f, 0.f, 0.f};
    // D[i][n] = (K @ E_sel)[i][n%3] : both 3x3 cam matrices in one matrix op.
    // All 8 waves execute it (EXEC all-ones requirement); only wave0 lanes store.
    Cm = __builtin_amdgcn_wmma_f32_16x16x4_f32(false, Am, false, Bm,
                                               (short)0, Cm, false, false);
    if (tid < 6) {
        lds_cam[tid * 3 + 0] = Cm[0];
        lds_cam[tid * 3 + 1] = Cm[1];
        lds_cam[tid * 3 + 2] = Cm[2];
    }
    __syncthreads();

    // lds_cam[j*3+i] = cam[i][j] (j<3: pred, j>=3: lab) -> row-major matrices
    float Mp[9], Ml[9];
    #pragma unroll
    for (int i = 0; i < 3; ++i)
        #pragma unroll
        for (int j = 0; j < 3; ++j) {
            Mp[i * 3 + j] = lds_cam[j * 3 + i];
            Ml[i * 3 + j] = lds_cam[(j + 3) * 3 + i];
        }
    float Pp[9], Pl[9];
    inv3(Mp, Pp);
    inv3(Ml, Pl);

    // Stream 1024 points: 2 points per 16B load, 2 loads per thread.
    const float4* Xb = (const float4*)(X + (size_t)b * (NPTS * 2));
    float acc = 0.0f;
    #pragma unroll
    for (int j = 0; j < 2; ++j) {
        float4 v = Xb[tid + j * 256];
        acc += point_term(v.x, v.y, Pp, Pl);
        acc += point_term(v.z, v.w, Pp, Pl);
    }

    // block reduction
    #pragma unroll
    for (int o = 16; o > 0; o >>= 1) acc += __shfl_down(acc, o, 32);
    if ((tid & 31) == 0) red[tid >> 5] = acc;
    __syncthreads();
    if (tid == 0) {
        float s = 0.0f;
        #pragma unroll
        for (int i = 0; i < 8; ++i) s += red[i];
        atomicAdd(out, s * (1.0f / NBATCH));
    }
}

extern "C" void kernel_launch(void* const* d_in, const int* in_sizes, int n_in,
                              void* d_out, int out_size, void* d_ws, size_t ws_size,
                              hipStream_t stream) {
    const float* X    = (const float*)d_in[0];   // (8192,1024,2)
    const float* y    = (const float*)d_in[1];   // (8192,7)
    const float* yp   = (const float*)d_in[2];   // (8192,5)
    const float* Kin  = (const float*)d_in[3];   // (3,3)
    float* out = (float*)d_out;                  // [loss, loss_t, loss_r]
    float* ws  = (float*)d_ws;                   // 8192*18 floats (E_pred|E_lab)

    gl_zero<<<1, 32, 0, stream>>>(out);
    gl_extrin<<<NBATCH / 256, 256, 0, stream>>>(y, yp, ws, out);
    gl_main<<<NBATCH, 256, 0, stream>>>(X, ws, Kin, out);
}